// DetectionLoss_79723182948415
// MI455X (gfx1250) — compile-verified
//
#include <hip/hip_runtime.h>
#include <math.h>

typedef __attribute__((ext_vector_type(2))) float v2f;
typedef __attribute__((ext_vector_type(8))) float v8f;

#define NCLS 11
#define GD 128
#define NANCH 3
#define NBATCH 16
#define NTGT 50
#define PLANE (GD*GD)                      // 16384
#define NCELLS (NBATCH*NANCH*GD*GD)        // 786432
#define K2_BLOCKS 1024
#define K2_THREADS 256
#define NTERMS 7

// scaled anchors = ANCHORS / (1024/128) = ANCHORS / 8
__constant__ float c_aw[3] = {14.5f, 19.5f, 46.625f};
__constant__ float c_ah[3] = {11.25f, 24.75f, 40.75f};

__device__ __forceinline__ float sigmoidf_(float v){ return 1.0f/(1.0f+expf(-v)); }

// ---------------- K0: init workspace ----------------
__global__ void k0_init(int* __restrict__ cellLast, int* __restrict__ cellCls,
                        int* __restrict__ counters){
    int idx = blockIdx.x*blockDim.x + threadIdx.x;
    int stride = gridDim.x*blockDim.x;
    for (int c = idx; c < NCELLS; c += stride){
        cellLast[c] = -1;
        cellCls[c]  = 0;
    }
    if (idx < 2) counters[idx] = 0;
}

// ---------------- K1: process 800 targets ----------------
__global__ void k1_targets(const float* __restrict__ x, const float* __restrict__ targets,
                           int* __restrict__ cellLast, int* __restrict__ cellCls,
                           float* __restrict__ tgtTab, int* __restrict__ counters){
    int tid = blockIdx.x*blockDim.x + threadIdx.x;
    if (tid >= NBATCH*NTGT) return;
    int b = tid / NTGT;
    const float* tg = targets + (size_t)tid*5;
    float t0 = tg[0], t1 = tg[1], t2 = tg[2], t3 = tg[3], t4 = tg[4];
    if (t0 + t1 + t2 + t3 + t4 == 0.0f) return;   // invalid target

    float gx = t1*(float)GD, gy = t2*(float)GD;
    float gw = t3*(float)GD, gh = t4*(float)GD;
    int gi = (int)gx; int gj = (int)gy;
    gi = gi < 0 ? 0 : (gi > GD-1 ? GD-1 : gi);
    gj = gj < 0 ? 0 : (gj > GD-1 ? GD-1 : gj);

    float iou[3]; bool over[3]; bool anyOver = false;
    #pragma unroll
    for (int a = 0; a < 3; a++){
        float inter = fminf(gw, c_aw[a]) * fminf(gh, c_ah[a]);
        float uni   = gw*gh + c_aw[a]*c_ah[a] - inter;
        iou[a] = inter / (uni + 1e-16f);
        over[a] = iou[a] > 0.3f;
        anyOver |= over[a];
    }
    int best = 0;                        // first-wins argmax (jnp.argmax semantics)
    if (iou[1] > iou[0]) best = 1;
    if (iou[2] > iou[best]) best = 2;
    bool sel[3];
    #pragma unroll
    for (int a = 0; a < 3; a++) sel[a] = anyOver ? over[a] : (a == best);

    int cls = (int)t0;
    tgtTab[tid*4+0] = gx - (float)gi;
    tgtTab[tid*4+1] = gy - (float)gj;
    tgtTab[tid*4+2] = gw;
    tgtTab[tid*4+3] = gh;

    bool corr = false;
    #pragma unroll
    for (int a = 0; a < 3; a++){
        if (!sel[a]) continue;
        int cidx = ((b*NANCH + a)*GD + gj)*GD + gi;
        atomicMax(&cellLast[cidx], tid);       // last-writer-wins (scan order == tid order)
        atomicOr(&cellCls[cidx], 1 << cls);    // class channels are never cleared -> union

        // pred_box at (b, a, gj, gi) for nCorrect
        const float* xb = x + ((size_t)(b*48 + a*16))*PLANE + (size_t)gj*GD + gi;
        float bx = sigmoidf_(xb[0*PLANE]) + (float)gi;
        float by = sigmoidf_(xb[1*PLANE]) + (float)gj;
        float bw = expf(xb[2*PLANE]) * c_aw[a];
        float bh = expf(xb[3*PLANE]) * c_ah[a];
        float iw = fminf(gx + gw*0.5f, bx + bw*0.5f) - fmaxf(gx - gw*0.5f, bx - bw*0.5f);
        float ih = fminf(gy + gh*0.5f, by + bh*0.5f) - fmaxf(gy - gh*0.5f, by - bh*0.5f);
        iw = fmaxf(iw, 0.0f); ih = fmaxf(ih, 0.0f);
        float inter = iw*ih;
        float uni = gw*gh + bw*bh - inter;
        if (inter/(uni + 1e-16f) > 0.5f) corr = true;
    }
    atomicAdd(&counters[0], 1);                // nGT
    if (corr) atomicAdd(&counters[1], 1);      // nCorrect
}

// ---------------- K2: streaming per-cell loss reduction ----------------
__global__ void k2_main(const float* __restrict__ x, const int* __restrict__ cellLast,
                        const int* __restrict__ cellCls, const float* __restrict__ tgtTab,
                        float* __restrict__ partials){
    __shared__ float sdata[K2_THREADS];
    float acc[NTERMS];
    #pragma unroll
    for (int t = 0; t < NTERMS; t++) acc[t] = 0.0f;

    int stride = gridDim.x*blockDim.x;
    for (int c = blockIdx.x*blockDim.x + threadIdx.x; c < NCELLS; c += stride){
        int i = c & (GD-1);
        int j = (c >> 7) & (GD-1);
        int a = (c >> 14) % NANCH;
        int b = c / (NANCH*PLANE);
        const float* xb = x + ((size_t)(b*48 + a*16))*PLANE + (size_t)j*GD + i;

        float conf = sigmoidf_(xb[4*PLANE]);
        int last = cellLast[c];
        if (last >= 0){
            // mask == tconf == 1 here
            acc[4] += (conf - 1.0f)*(conf - 1.0f);
            float px = sigmoidf_(xb[0*PLANE]);
            float py = sigmoidf_(xb[1*PLANE]);
            float pw = xb[2*PLANE];
            float ph = xb[3*PLANE];
            float fx = tgtTab[last*4+0], fy = tgtTab[last*4+1];
            float gw = tgtTab[last*4+2], gh = tgtTab[last*4+3];
            float tw = logf(gw / c_aw[a] + 1e-16f);
            float th = logf(gh / c_ah[a] + 1e-16f);
            acc[0] += (px - fx)*(px - fx);
            acc[1] += (py - fy)*(py - fy);
            acc[2] += (pw - tw)*(pw - tw);
            acc[3] += (ph - th)*(ph - th);

            // stable softmax over 11 class logits
            float l[NCLS];
            l[0] = xb[5*PLANE];
            float m = l[0];
            #pragma unroll
            for (int k = 1; k < NCLS; k++){ l[k] = xb[(5+k)*PLANE]; m = fmaxf(m, l[k]); }
            float Z = 0.0f;
            #pragma unroll
            for (int k = 0; k < NCLS; k++){ l[k] = expf(l[k] - m); Z += l[k]; }
            float invZ = 1.0f / Z;
            int cm = cellCls[c];
            float lc = 0.0f;
            #pragma unroll
            for (int k = 0; k < NCLS; k++){
                float p  = l[k]*invZ;
                float lp = fmaxf(logf(p), -100.0f);
                float l1 = fmaxf(logf(1.0f - p), -100.0f);
                float tt = (float)((cm >> k) & 1);
                lc += tt*lp + (1.0f - tt)*l1;
            }
            acc[6] -= lc;
        } else {
            acc[5] += conf*conf;   // noobj term (pre-0.5 scale)
        }
    }

    // deterministic fixed-order block reduction, one term at a time
    for (int t = 0; t < NTERMS; t++){
        sdata[threadIdx.x] = acc[t];
        __syncthreads();
        for (int s = K2_THREADS/2; s > 0; s >>= 1){
            if (threadIdx.x < s) sdata[threadIdx.x] += sdata[threadIdx.x + s];
            __syncthreads();
        }
        if (threadIdx.x == 0) partials[t*K2_BLOCKS + blockIdx.x] = sdata[0];
        __syncthreads();
    }
}

// ---------------- K3: final WMMA ones-vector reduction (1 wave) ----------------
__global__ void k3_final(const float* __restrict__ partials, const int* __restrict__ counters,
                         float* __restrict__ out){
    int lane = threadIdx.x;           // 32 lanes, EXEC all ones for WMMA
    v2f bones; bones.x = 1.0f; bones.y = 1.0f;   // B = 4x16 all-ones
    float sums[NTERMS];
    for (int t = 0; t < NTERMS; t++){
        v8f accv = {0.f,0.f,0.f,0.f,0.f,0.f,0.f,0.f};
        const float* p = partials + t*K2_BLOCKS;
        for (int base = 0; base < K2_BLOCKS; base += 64){
            v2f av;                                   // A = 16x4, 64 partials per issue
            av.x = p[base + lane*2 + 0];
            av.y = p[base + lane*2 + 1];
            // D[m][n] = sum_k A[m][k] + C[m][n]
            accv = __builtin_amdgcn_wmma_f32_16x16x4_f32(
                false, av, false, bones, (short)0, accv, false, false);
        }
        // column N=0 lives in lane 0 (M=0..7) and lane 16 (M=8..15)
        float s = accv[0]+accv[1]+accv[2]+accv[3]+accv[4]+accv[5]+accv[6]+accv[7];
        sums[t] = __shfl(s, 0, 32) + __shfl(s, 16, 32);
    }
    if (lane == 0){
        float lx = 100.0f * sums[0];
        float ly = 100.0f * sums[1];
        float lw = 100.0f * sums[2];
        float lh = 100.0f * sums[3];
        float lconf = sums[4];
        float lcn = 0.5f * sums[5];
        float lcls = sums[6];
        float loss = lx + ly + lw + lh + lconf + lcls + lcn;
        int nGT = counters[0], nC = counters[1];
        float recall = (nGT > 0) ? ((float)nC / (float)(nGT < 1 ? 1 : nGT)) : 1.0f;
        out[0] = loss; out[1] = lx; out[2] = ly; out[3] = lw; out[4] = lh;
        out[5] = lconf; out[6] = lcn; out[7] = lcls; out[8] = recall;
    }
}

extern "C" void kernel_launch(void* const* d_in, const int* in_sizes, int n_in,
                              void* d_out, int out_size, void* d_ws, size_t ws_size,
                              hipStream_t stream){
    const float* x       = (const float*)d_in[0];   // (16,48,128,128) f32
    const float* targets = (const float*)d_in[1];   // (16,50,5) f32

    char* ws = (char*)d_ws;
    size_t off = 0;
    int*   cellLast = (int*)(ws + off);  off += (size_t)NCELLS*sizeof(int);      // 3 MB
    int*   cellCls  = (int*)(ws + off);  off += (size_t)NCELLS*sizeof(int);      // 3 MB
    float* tgtTab   = (float*)(ws + off); off += (size_t)NBATCH*NTGT*4*sizeof(float);
    int*   counters = (int*)(ws + off);  off += 64;                               // aligned
    float* partials = (float*)(ws + off);                                         // 7*1024 f32

    k0_init<<<512, 256, 0, stream>>>(cellLast, cellCls, counters);
    k1_targets<<<(NBATCH*NTGT + 255)/256, 256, 0, stream>>>(x, targets, cellLast, cellCls,
                                                            tgtTab, counters);
    k2_main<<<K2_BLOCKS, K2_THREADS, 0, stream>>>(x, cellLast, cellCls, tgtTab, partials);
    k3_final<<<1, 32, 0, stream>>>(partials, counters, (float*)d_out);
}